// GodSlayer_25271587569863
// MI455X (gfx1250) — compile-verified
//
#include <hip/hip_runtime.h>
#include <math.h>

typedef __attribute__((ext_vector_type(2))) float v2f;
typedef __attribute__((ext_vector_type(8))) float v8f;

#define Bsz   16
#define Lsz   512
#define Csz   4
#define NL    4
#define NNB   2
#define HIDsz 256
#define LMASK 511
#define NCHUNK 8            // i-chunks for diag partial sums (64 rows each)

__device__ __forceinline__ float celu1(float v) {
    return v > 0.0f ? v : expm1f(v);
}

// ---------------------------------------------------------------------------
// Kernel 1: Gram matrix g = n . n^T per batch via v_wmma_f32_16x16x4_f32.
// One wave per 16x16 tile; K=3 padded to 4 with zeros. Diagonal zeroed.
// Result broadcast into all C=4 channels of X (B,C,L,L).
// ---------------------------------------------------------------------------
__global__ void k_gram(const float* __restrict__ n, float* __restrict__ X) {
    const int lane = threadIdx.x & 31;
    const int wid  = (blockIdx.x * blockDim.x + threadIdx.x) >> 5;
    const int b    = wid >> 10;          // 1024 tiles per batch (32x32 tiles)
    const int tile = wid & 1023;
    const int i0   = (tile >> 5) << 4;
    const int j0   = (tile & 31) << 4;

    const int half = lane >> 4;          // half 0: K=0,1 ; half 1: K=2,3
    const int l16  = lane & 15;

    const float* nb = n + (size_t)b * Lsz * 3;

    v2f a, bb;
    {   // A tile (16x4): lane half selects K pair; row = i0 + l16
        const float* row = nb + (size_t)(i0 + l16) * 3;
        a.x = half ? row[2] : row[0];
        a.y = half ? 0.0f   : row[1];
    }
    {   // B tile (4x16) = n^T: column = j0 + l16
        const float* col = nb + (size_t)(j0 + l16) * 3;
        bb.x = half ? col[2] : col[0];
        bb.y = half ? 0.0f   : col[1];
    }

    v8f acc = {};
    acc = __builtin_amdgcn_wmma_f32_16x16x4_f32(false, a, false, bb,
                                                (short)0, acc, false, false);

    // D layout: VGPR r -> M = r + 8*half (per half), N = j0 + l16
    const int j = j0 + l16;
    #pragma unroll
    for (int r = 0; r < 8; ++r) {
        const int i = i0 + r + (half << 3);
        float v = acc[r];
        if (i == j) v = 0.0f;
        const size_t off = (size_t)i * Lsz + j;
        #pragma unroll
        for (int c = 0; c < Csz; ++c) {
            X[((size_t)(b * Csz + c) * Lsz) * Lsz + off] = v;
        }
    }
}

// ---------------------------------------------------------------------------
// Kernel 2: one residual stencil layer, 4 columns per thread via b128.
// x symmetric at entry => symmetrization is a no-op; just mask the diagonal.
//   y   = w_self*x + sum_d w_nbr[d]*(x[i,j-d]+x[i,j+d]+x[i-d,j]+x[i+d,j])
//   out = (i==j) ? 0 : x + celu(y)
// All halo data comes from aligned float4 groups (L=512 divisible by 4, so
// group-aligned cyclic wraparound is exact).
// ---------------------------------------------------------------------------
__global__ void k_layer(const float* __restrict__ Xin, float* __restrict__ Xout,
                        const float* __restrict__ w_self,
                        const float* __restrict__ w_nbr, int ell) {
    const size_t idx = (size_t)blockIdx.x * blockDim.x + threadIdx.x;
    const int g  = (int)(idx & 127);                 // 4-col group in row
    const int i  = (int)((idx >> 7) & LMASK);
    const int c  = (int)((idx >> 16) & 3);
    const size_t mbase = (idx >> 16) * ((size_t)Lsz * Lsz);  // (b*C + c)*L*L
    const float* M = Xin + mbase;
    const int j0 = g << 2;

    const float ws = w_self[c * NL + ell];
    const float w1 = w_nbr[(c * NL + ell) * NNB + 0];
    const float w2 = w_nbr[(c * NL + ell) * NNB + 1];

    const float4 ctr = *(const float4*)(M + (size_t)i * Lsz + j0);
    const float4 lft = *(const float4*)(M + (size_t)i * Lsz + ((j0 - 4) & LMASK));
    const float4 rgt = *(const float4*)(M + (size_t)i * Lsz + ((j0 + 4) & LMASK));
    const float4 up1 = *(const float4*)(M + (size_t)((i - 1) & LMASK) * Lsz + j0);
    const float4 dn1 = *(const float4*)(M + (size_t)((i + 1) & LMASK) * Lsz + j0);
    const float4 up2 = *(const float4*)(M + (size_t)((i - 2) & LMASK) * Lsz + j0);
    const float4 dn2 = *(const float4*)(M + (size_t)((i + 2) & LMASK) * Lsz + j0);

    const float cc[4] = {ctr.x, ctr.y, ctr.z, ctr.w};
    const float ll[4] = {lft.x, lft.y, lft.z, lft.w};
    const float rr[4] = {rgt.x, rgt.y, rgt.z, rgt.w};
    const float u1[4] = {up1.x, up1.y, up1.z, up1.w};
    const float d1[4] = {dn1.x, dn1.y, dn1.z, dn1.w};
    const float u2[4] = {up2.x, up2.y, up2.z, up2.w};
    const float d2[4] = {dn2.x, dn2.y, dn2.z, dn2.w};

    float4 o;
    float* op = (float*)&o;
    #pragma unroll
    for (int t = 0; t < 4; ++t) {
        const float x   = cc[t];
        const float jm1 = (t >= 1) ? cc[t - 1] : ll[3];
        const float jp1 = (t <= 2) ? cc[t + 1] : rr[0];
        const float jm2 = (t >= 2) ? cc[t - 2] : ll[2 + t];
        const float jp2 = (t <= 1) ? cc[t + 2] : rr[t - 2];
        float y = ws * x;
        y += w1 * (jm1 + jp1 + u1[t] + d1[t]);   // ref roll order: j-d, j+d, i-d, i+d
        y += w2 * (jm2 + jp2 + u2[t] + d2[t]);
        const float v = x + celu1(y);
        op[t] = ((j0 + t) == i) ? 0.0f : v;
    }
    *(float4*)(Xout + mbase + (size_t)i * Lsz + j0) = o;
}

// ---------------------------------------------------------------------------
// Kernel 3a: partial cyclic diagonal sums over 64-row i-chunks.
// part[bc,ch,r] = sum_{i in chunk} x[bc,i,(i+r)%L]
// Lanes own consecutive r -> coalesced reads; register acc (deterministic).
// ---------------------------------------------------------------------------
__global__ void k_diag_partial(const float* __restrict__ X, float* __restrict__ part) {
    const int bc = blockIdx.x >> 3;                  // 0..B*C-1
    const int ch = blockIdx.x & (NCHUNK - 1);
    const float* M = X + (size_t)bc * Lsz * Lsz;
    const int i0 = ch * (Lsz / NCHUNK);
    for (int r = threadIdx.x; r < Lsz; r += blockDim.x) {
        float acc = 0.0f;
        for (int i = i0; i < i0 + (Lsz / NCHUNK); ++i) {
            acc += M[(size_t)i * Lsz + ((i + r) & LMASK)];
        }
        part[((size_t)bc * NCHUNK + ch) * Lsz + r] = acc;
    }
}

// Kernel 3b: reduce chunk partials -> sep[b,c,r] (fixed order, deterministic)
__global__ void k_diag_reduce(const float* __restrict__ part, float* __restrict__ sep) {
    const int bc = blockIdx.x;
    for (int r = threadIdx.x; r < Lsz; r += blockDim.x) {
        float acc = 0.0f;
        #pragma unroll
        for (int ch = 0; ch < NCHUNK; ++ch) {
            acc += part[((size_t)bc * NCHUNK + ch) * Lsz + r];
        }
        sep[(size_t)bc * Lsz + r] = acc;
    }
}

// ---------------------------------------------------------------------------
// Kernel 4: h = celu(sep_flat @ W1 + b1), M=16, K=2048, N=256.
// One wave per 16-column tile; 512 chained v_wmma_f32_16x16x4_f32.
// ---------------------------------------------------------------------------
__global__ void k_mlp1(const float* __restrict__ sep, const float* __restrict__ W1,
                       const float* __restrict__ b1, float* __restrict__ h) {
    const int lane = threadIdx.x & 31;
    const int wid  = (blockIdx.x * blockDim.x + threadIdx.x) >> 5;  // 0..15
    const int n0   = wid << 4;
    const int half = lane >> 4;
    const int l16  = lane & 15;
    const int K    = Csz * Lsz;                      // 2048

    v8f acc = {};
    const int nc = n0 + l16;
    for (int k0 = 0; k0 < K; k0 += 4) {
        const int ka = k0 + (half << 1);             // half-dependent K pair
        v2f a, bb;
        a.x  = sep[l16 * K + ka];
        a.y  = sep[l16 * K + ka + 1];
        bb.x = W1[(size_t)ka       * HIDsz + nc];
        bb.y = W1[(size_t)(ka + 1) * HIDsz + nc];
        acc = __builtin_amdgcn_wmma_f32_16x16x4_f32(false, a, false, bb,
                                                    (short)0, acc, false, false);
    }
    const float bias = b1[nc];
    #pragma unroll
    for (int r = 0; r < 8; ++r) {
        const int m = r + (half << 3);
        h[m * HIDsz + nc] = celu1(acc[r] + bias);
    }
}

// ---------------------------------------------------------------------------
// Kernel 5: y = h @ W2 + b2 ; out = exp(-y)
// ---------------------------------------------------------------------------
__global__ void k_final(const float* __restrict__ h, const float* __restrict__ W2,
                        const float* __restrict__ b2, float* __restrict__ out) {
    const int t = threadIdx.x;
    if (t < Bsz) {
        float acc = 0.0f;
        for (int k = 0; k < HIDsz; ++k) acc += h[t * HIDsz + k] * W2[k];
        out[t] = expf(-(acc + b2[0]));
    }
}

extern "C" void kernel_launch(void* const* d_in, const int* in_sizes, int n_in,
                              void* d_out, int out_size, void* d_ws, size_t ws_size,
                              hipStream_t stream) {
    const float* n      = (const float*)d_in[0];
    const float* w_self = (const float*)d_in[1];
    const float* w_nbr  = (const float*)d_in[2];
    const float* W1     = (const float*)d_in[3];
    const float* b1     = (const float*)d_in[4];
    const float* W2     = (const float*)d_in[5];
    const float* b2     = (const float*)d_in[6];
    float* out = (float*)d_out;

    const size_t matBytes = (size_t)Bsz * Csz * Lsz * Lsz * sizeof(float); // 64 MiB
    const size_t sepBytes = (size_t)Bsz * Csz * Lsz * sizeof(float);       // 128 KiB
    const size_t hBytes   = (size_t)Bsz * HIDsz * sizeof(float);           // 16 KiB
    char* ws = (char*)d_ws;
    float* XA   = (float*)(ws);
    float* XB   = (float*)(ws + matBytes);
    float* sep  = (float*)(ws + 2 * matBytes);
    float* h    = (float*)(ws + 2 * matBytes + sepBytes);
    float* part = (float*)(ws + 2 * matBytes + sepBytes + hBytes);         // 1 MiB
    float* bufs[2] = {XA, XB};

    // 1) Gram init (WMMA), broadcast to 4 channels: 16384 waves
    k_gram<<<2048, 256, 0, stream>>>(n, XA);

    // 2) 4 stencil layers, ping-pong XA <-> XB (ends back in XA)
    for (int ell = 0; ell < NL; ++ell) {
        k_layer<<<(Bsz * Csz * Lsz * Lsz / 4) / 256, 256, 0, stream>>>(
            bufs[ell & 1], bufs[(ell + 1) & 1], w_self, w_nbr, ell);
    }

    // 3) cyclic diagonal sums: partial over i-chunks, then reduce
    k_diag_partial<<<Bsz * Csz * NCHUNK, 256, 0, stream>>>(XA, part);
    k_diag_reduce<<<Bsz * Csz, 256, 0, stream>>>(part, sep);

    // 4) MLP hidden layer (WMMA): 16 waves
    k_mlp1<<<2, 256, 0, stream>>>(sep, W1, b1, h);

    // 5) head + exp(-y)
    k_final<<<1, 32, 0, stream>>>(h, W2, b2, out);
}